// TransformerBlock_44255343018256
// MI455X (gfx1250) — compile-verified
//
#include <hip/hip_runtime.h>

typedef __attribute__((ext_vector_type(2))) float v2f;
typedef __attribute__((ext_vector_type(8))) float v8f;

// fp32 WMMA: D(16x16) = A(16x4) * B(4x16) + C   (wave32, full-precision path)
__device__ __forceinline__ v8f wmma4(v2f a, v2f b, v8f c) {
  return __builtin_amdgcn_wmma_f32_16x16x4_f32(
      /*neg_a=*/false, a, /*neg_b=*/false, b,
      /*c_mod=*/(short)0, c, /*reuse_a=*/false, /*reuse_b=*/false);
}

__device__ __forceinline__ float gelu_exact(float x) {
  return 0.5f * x * (1.0f + erff(x * 0.70710678118654752440f));
}

// ---------------------------------------------------------------------------
// LayerNorm: one block per row of D=1024. Biased variance (matches jnp.var).
// ---------------------------------------------------------------------------
__global__ __launch_bounds__(256) void ln_kernel(const float* __restrict__ x,
                                                 const float* __restrict__ gw,
                                                 const float* __restrict__ bw,
                                                 float* __restrict__ out) {
  const int row  = blockIdx.x;
  const int t    = threadIdx.x;
  const int wave = t >> 5, lane = t & 31;

  const float4 v = ((const float4*)(x + (size_t)row * 1024))[t];
  float s  = v.x + v.y + v.z + v.w;
  float s2 = v.x * v.x + v.y * v.y + v.z * v.z + v.w * v.w;
#pragma unroll
  for (int off = 16; off >= 1; off >>= 1) {
    s  += __shfl_xor(s, off, 32);
    s2 += __shfl_xor(s2, off, 32);
  }
  __shared__ float red[16];
  if (lane == 0) { red[2 * wave] = s; red[2 * wave + 1] = s2; }
  __syncthreads();
  float S1 = 0.f, S2 = 0.f;
#pragma unroll
  for (int w = 0; w < 8; ++w) { S1 += red[2 * w]; S2 += red[2 * w + 1]; }
  const float mean = S1 * (1.0f / 1024.0f);
  const float var  = S2 * (1.0f / 1024.0f) - mean * mean;
  const float rstd = rsqrtf(var + 1e-5f);

  const float4 g4 = ((const float4*)gw)[t];
  const float4 b4 = ((const float4*)bw)[t];
  float4 o;
  o.x = (v.x - mean) * rstd * g4.x + b4.x;
  o.y = (v.y - mean) * rstd * g4.y + b4.y;
  o.z = (v.z - mean) * rstd * g4.z + b4.z;
  o.w = (v.w - mean) * rstd * g4.w + b4.w;
  ((float4*)(out + (size_t)row * 1024))[t] = o;
}

// ---------------------------------------------------------------------------
// GEMM: C[M,N] = A[M,K] @ W[K,N] + bias (+ residual | gelu).
// 8 waves/block; each wave: 32x32 output = 2x2 WMMA tiles; block: 128x64.
// EPI: 0 = bias, 1 = bias+residual, 2 = bias+gelu
// ---------------------------------------------------------------------------
template <int EPI>
__global__ __launch_bounds__(256) void gemm_kernel(const float* __restrict__ A,
                                                   const float* __restrict__ W,
                                                   const float* __restrict__ bias,
                                                   const float* __restrict__ res,
                                                   float* __restrict__ C,
                                                   int N, int K) {
  const int lane  = threadIdx.x & 31;
  const int wave  = threadIdx.x >> 5;
  const int row   = lane & 15;   // M (A frag) / N (B,C frags) position
  const int kh    = lane >> 4;   // K-half selector (0 or 1)
  const int waveM = wave >> 1;   // 0..3
  const int waveN = wave & 1;    // 0..1
  const int m0 = blockIdx.x * 128 + waveM * 32;
  const int n0 = blockIdx.y * 64 + waveN * 32;

  v8f c00 = {}, c01 = {}, c10 = {}, c11 = {};

  const float* a0p = A + (m0 + row) * K + 2 * kh;
  const float* a1p = a0p + 16 * K;
  const float* bp  = W + (2 * kh) * N + n0 + row;

#pragma unroll 4
  for (int k = 0; k < K; k += 4) {
    v2f a0 = *(const v2f*)a0p;
    v2f a1 = *(const v2f*)a1p;
    v2f b0, b1;
    b0.x = bp[0];  b0.y = bp[N];
    b1.x = bp[16]; b1.y = bp[N + 16];
    c00 = wmma4(a0, b0, c00);
    c01 = wmma4(a0, b1, c01);
    c10 = wmma4(a1, b0, c10);
    c11 = wmma4(a1, b1, c11);
    a0p += 4; a1p += 4; bp += 4 * N;
  }

  const float bv0 = bias[n0 + row];
  const float bv1 = bias[n0 + 16 + row];
#pragma unroll
  for (int i = 0; i < 8; ++i) {
    const int r0 = m0 + i + 8 * kh;       // tiles (m0, *)
    const int r1 = r0 + 16;               // tiles (m0+16, *)
    const int cA = n0 + row;              // tiles (*, n0)
    const int cB = cA + 16;               // tiles (*, n0+16)
    float v00 = c00[i] + bv0, v01 = c01[i] + bv1;
    float v10 = c10[i] + bv0, v11 = c11[i] + bv1;
    if (EPI == 1) {
      v00 += res[(size_t)r0 * N + cA]; v01 += res[(size_t)r0 * N + cB];
      v10 += res[(size_t)r1 * N + cA]; v11 += res[(size_t)r1 * N + cB];
    }
    if (EPI == 2) {
      v00 = gelu_exact(v00); v01 = gelu_exact(v01);
      v10 = gelu_exact(v10); v11 = gelu_exact(v11);
    }
    C[(size_t)r0 * N + cA] = v00; C[(size_t)r0 * N + cB] = v01;
    C[(size_t)r1 * N + cA] = v10; C[(size_t)r1 * N + cB] = v11;
  }
}

// ---------------------------------------------------------------------------
// Flash attention, fp32 WMMA. qkv layout: [B,S,3,H,Hd], row stride 3*D=3072.
// 4 waves/block; wave = one 16-row Q tile of one (b,h).
// ---------------------------------------------------------------------------
__global__ __launch_bounds__(128) void attn_kernel(const float* __restrict__ qkv,
                                                   float* __restrict__ out) {
  constexpr int S = 2048, D = 1024, R3D = 3072;
  const int lane = threadIdx.x & 31;
  const int wave = threadIdx.x >> 5;
  const int row  = lane & 15;
  const int kh   = lane >> 4;
  const int bh   = blockIdx.x;            // 0..31
  const int b    = bh >> 4, h = bh & 15;
  const int q0   = (blockIdx.y * 4 + wave) * 16;

  const float* qbase = qkv + (size_t)b * S * R3D + h * 64;  // c=0
  const float* kbase = qbase + D;                            // c=1
  const float* vbase = qbase + 2 * D;                        // c=2

  // Preload Q tile (16x64) as 16 A-fragments.
  v2f qf[16];
  {
    const float* qrow = qbase + (size_t)(q0 + row) * R3D + 2 * kh;
#pragma unroll
    for (int j = 0; j < 16; ++j) qf[j] = *(const v2f*)(qrow + 4 * j);
  }

  v8f o0 = {}, o1 = {}, o2 = {}, o3 = {};
  float mrun[8], lrun[8];
#pragma unroll
  for (int i = 0; i < 8; ++i) { mrun[i] = -1e30f; lrun[i] = 0.f; }

  __shared__ float Pbuf[4][16 * 18];      // stride 18 -> conflict-free
  float* P = Pbuf[wave];

  for (int kt = 0; kt < S; kt += 16) {
    // ---- scores (16x16) = Q @ K^T : 16 WMMAs over K=64 ----
    v8f sc = {};
    {
      const float* krow = kbase + (size_t)(kt + row) * R3D + 2 * kh;
#pragma unroll
      for (int j = 0; j < 16; ++j) {
        v2f kf = *(const v2f*)(krow + 4 * j);
        sc = wmma4(qf[j], kf, sc);
      }
    }
    // ---- online softmax; write P to LDS in A-layout-friendly form ----
    float alpha[8];
#pragma unroll
    for (int i = 0; i < 8; ++i) {
      float s = sc[i] * 0.125f;           // Hd^-0.5
      float mx = s;
      mx = fmaxf(mx, __shfl_xor(mx, 1, 32));
      mx = fmaxf(mx, __shfl_xor(mx, 2, 32));
      mx = fmaxf(mx, __shfl_xor(mx, 4, 32));
      mx = fmaxf(mx, __shfl_xor(mx, 8, 32));
      const float mnew = fmaxf(mrun[i], mx);
      const float a    = __expf(mrun[i] - mnew);
      const float p    = __expf(s - mnew);
      float sum = p;
      sum += __shfl_xor(sum, 1, 32);
      sum += __shfl_xor(sum, 2, 32);
      sum += __shfl_xor(sum, 4, 32);
      sum += __shfl_xor(sum, 8, 32);
      lrun[i] = lrun[i] * a + sum;
      mrun[i] = mnew;
      alpha[i] = a;
      P[(i + 8 * kh) * 18 + row] = p;     // C-layout -> LDS (row, col)
    }
#pragma unroll
    for (int i = 0; i < 8; ++i) {
      o0[i] *= alpha[i]; o1[i] *= alpha[i];
      o2[i] *= alpha[i]; o3[i] *= alpha[i];
    }
    // ---- O += P(16x16) @ V(16x64) : 16 WMMAs ----
#pragma unroll
    for (int j = 0; j < 4; ++j) {
      v2f pa = *(const v2f*)(P + row * 18 + 4 * j + 2 * kh);  // A frag from LDS
      const float* vr0 = vbase + (size_t)(kt + 4 * j + 2 * kh) * R3D;
      const float* vr1 = vr0 + R3D;
      v2f vb0 = { vr0[row],      vr1[row]      };
      v2f vb1 = { vr0[16 + row], vr1[16 + row] };
      v2f vb2 = { vr0[32 + row], vr1[32 + row] };
      v2f vb3 = { vr0[48 + row], vr1[48 + row] };
      o0 = wmma4(pa, vb0, o0);
      o1 = wmma4(pa, vb1, o1);
      o2 = wmma4(pa, vb2, o2);
      o3 = wmma4(pa, vb3, o3);
    }
  }

  // ---- epilogue: out[b, q, h*64 + d] = O / l ----
#pragma unroll
  for (int i = 0; i < 8; ++i) {
    const int qr = q0 + i + 8 * kh;
    const float inv = 1.0f / lrun[i];
    float* orow = out + ((size_t)b * S + qr) * D + h * 64 + row;
    orow[0]  = o0[i] * inv;
    orow[16] = o1[i] * inv;
    orow[32] = o2[i] * inv;
    orow[48] = o3[i] * inv;
  }
}

// ---------------------------------------------------------------------------
// Host-side orchestration.
// ---------------------------------------------------------------------------
extern "C" void kernel_launch(void* const* d_in, const int* in_sizes, int n_in,
                              void* d_out, int out_size, void* d_ws, size_t ws_size,
                              hipStream_t stream) {
  const float* x      = (const float*)d_in[0];
  const float* ln1_g  = (const float*)d_in[1];
  const float* ln1_b  = (const float*)d_in[2];
  const float* qkv_w  = (const float*)d_in[3];
  const float* qkv_b  = (const float*)d_in[4];
  const float* proj_w = (const float*)d_in[5];
  const float* proj_b = (const float*)d_in[6];
  const float* ln2_g  = (const float*)d_in[7];
  const float* ln2_b  = (const float*)d_in[8];
  const float* fc1_w  = (const float*)d_in[9];
  const float* fc1_b  = (const float*)d_in[10];
  const float* fc2_w  = (const float*)d_in[11];
  const float* fc2_b  = (const float*)d_in[12];
  float* out = (float*)d_out;

  const int M = 4096;  // B*S
  float* bufA = (float*)d_ws;                       // 4096*4096: h, then hmid
  float* bufB = bufA + (size_t)4096 * 4096;         // 4096*3072: qkv, then h2
  float* bufC = bufB + (size_t)4096 * 3072;         // 4096*1024: attn out
  float* bufD = bufC + (size_t)4096 * 1024;         // 4096*1024: x1

  // h = LN1(x)
  ln_kernel<<<M, 256, 0, stream>>>(x, ln1_g, ln1_b, bufA);
  // qkv = h @ qkv_w + qkv_b
  gemm_kernel<0><<<dim3(M / 128, 3072 / 64), 256, 0, stream>>>(
      bufA, qkv_w, qkv_b, nullptr, bufB, 3072, 1024);
  // attention
  attn_kernel<<<dim3(32, 32), 128, 0, stream>>>(bufB, bufC);
  // x1 = x + attn @ proj_w + proj_b
  gemm_kernel<1><<<dim3(M / 128, 1024 / 64), 256, 0, stream>>>(
      bufC, proj_w, proj_b, x, bufD, 1024, 1024);
  // h2 = LN2(x1)
  ln_kernel<<<M, 256, 0, stream>>>(bufD, ln2_g, ln2_b, bufB);
  // hmid = gelu(h2 @ fc1_w + fc1_b)
  gemm_kernel<2><<<dim3(M / 128, 4096 / 64), 256, 0, stream>>>(
      bufB, fc1_w, fc1_b, nullptr, bufA, 4096, 1024);
  // out = x1 + hmid @ fc2_w + fc2_b
  gemm_kernel<1><<<dim3(M / 128, 1024 / 64), 256, 0, stream>>>(
      bufA, fc2_w, fc2_b, bufD, out, 1024, 4096);
}